// ViewTransformerLSSBEVDepthForward_15814069584340
// MI455X (gfx1250) — compile-verified
//
#include <hip/hip_runtime.h>
#include <math.h>

typedef float v2f __attribute__((ext_vector_type(2)));
typedef float v8f __attribute__((ext_vector_type(8)));

#define NCAM   6
#define CCH    80
#define DBINS  112
#define HFD    16
#define WFD    44
#define NXV    128
#define NYV    128
#define HW     (HFD * WFD)                 // 704
#define NPTS   (NCAM * DBINS * HW)         // 473088
#define MAT_STRIDE 24                      // floats per camera in ws

// ---------------------------------------------------------------------------
// 3x3 inverse via adjugate
// ---------------------------------------------------------------------------
__device__ inline void inv3x3(const float* m, float* o) {
    float a = m[0], b = m[1], c = m[2];
    float d = m[3], e = m[4], f = m[5];
    float g = m[6], h = m[7], i = m[8];
    float A =  e * i - f * h;
    float B = -(d * i - f * g);
    float C =  d * h - e * g;
    float det = a * A + b * B + c * C;
    float id = 1.0f / det;
    o[0] = A * id;              o[1] = -(b * i - c * h) * id;  o[2] =  (b * f - c * e) * id;
    o[3] = B * id;              o[4] =  (a * i - c * g) * id;  o[5] = -(a * f - c * d) * id;
    o[6] = C * id;              o[7] = -(a * h - b * g) * id;  o[8] =  (a * e - b * d) * id;
}

// ---------------------------------------------------------------------------
// Per-camera matrix prep: ws layout per camera n (24 floats):
//   [0..8]  inv(post_rots)   [9..17] combine = rots @ inv(intrins)
//   [18..20] trans           [21..23] post_trans
// ---------------------------------------------------------------------------
__global__ void prep_kernel(const float* __restrict__ rots,
                            const float* __restrict__ trans,
                            const float* __restrict__ intrins,
                            const float* __restrict__ post_rots,
                            const float* __restrict__ post_trans,
                            float* __restrict__ mats) {
    int n = threadIdx.x;
    if (n >= NCAM) return;
    float ipr[9], iin[9], cmb[9];
    inv3x3(post_rots + n * 9, ipr);
    inv3x3(intrins + n * 9, iin);
    const float* R = rots + n * 9;
    for (int i = 0; i < 3; ++i)
        for (int j = 0; j < 3; ++j) {
            float s = 0.0f;
            for (int k = 0; k < 3; ++k) s += R[i * 3 + k] * iin[k * 3 + j];
            cmb[i * 3 + j] = s;
        }
    float* o = mats + n * MAT_STRIDE;
    for (int k = 0; k < 9; ++k) { o[k] = ipr[k]; o[9 + k] = cmb[k]; }
    for (int k = 0; k < 3; ++k) { o[18 + k] = trans[n * 3 + k]; o[21 + k] = post_trans[n * 3 + k]; }
}

// ---------------------------------------------------------------------------
// Geometry: 16 frustum points per wave; final affine transform (combine, trans)
// done with one V_WMMA_F32_16X16X4_F32 per wave (D = A[16x4] * B[4x16]).
// A layout (ISA 7.12.2): lane L holds row M=L&15; VGPR0=K{0|2}, VGPR1=K{1|3}.
// Every lane computes q for point (lane&15), so A needs no cross-lane moves:
//   lanes 0-15 carry (qx,qy), lanes 16-31 carry (qz,1).
// D layout: VGPR r, lanes 0-15 -> (M=r, N=lane); lanes 16-31 -> (M=r+8).
// ---------------------------------------------------------------------------
__global__ void geom_kernel(const float* __restrict__ mats,
                            int* __restrict__ voxIdx) {
    const int lane = threadIdx.x & 31;
    const int wave = blockIdx.x * (blockDim.x >> 5) + (threadIdx.x >> 5);
    const int pointBase = wave * 16;
    const int m16 = lane & 15;
    const bool hi = lane >= 16;

    int p = pointBase + m16;
    int w = p % WFD;  int t = p / WFD;
    int h = t % HFD;  t /= HFD;
    int d = t % DBINS;
    int n = t / DBINS;                       // uniform across the wave

    const float* M = mats + n * MAT_STRIDE;

    // frustum point (augmented-image coords)
    float xp  = (float)w * (703.0f / 43.0f); // linspace(0, 703, 44)
    float yp  = (float)h * 17.0f;            // linspace(0, 255, 16)
    float dep = 2.0f + 0.5f * (float)d;      // arange(2, 58, 0.5)

    // q = inv(post_rots) @ (fr - post_trans)
    float qx = xp - M[21], qy = yp - M[22], qz = dep - M[23];
    float r0 = M[0] * qx + M[1] * qy + M[2] * qz;
    float r1 = M[3] * qx + M[4] * qy + M[5] * qz;
    float r2 = M[6] * qx + M[7] * qy + M[8] * qz;
    float sx = r0 * r2, sy = r1 * r2, sz = r2;   // (x*d, y*d, d)

    // A matrix operand
    v2f A;
    A.x = hi ? sz : sx;
    A.y = hi ? 1.0f : sy;

    // B matrix operand: B[k][j] = combine[j][k] (k<3), B[3][j] = trans[j], j<3
    float b0 = 0.0f, b1 = 0.0f;
    if (m16 < 3) {
        const float* C = M + 9;
        b0 = hi ? C[m16 * 3 + 2] : C[m16 * 3 + 0];
        b1 = hi ? M[18 + m16]    : C[m16 * 3 + 1];
    }
    v2f Bm; Bm.x = b0; Bm.y = b1;

    v8f acc = {};
    acc = __builtin_amdgcn_wmma_f32_16x16x4_f32(
        /*neg_a=*/false, A, /*neg_b=*/false, Bm,
        /*c_mod=*/(short)0, acc, /*reuse_a=*/false, /*reuse_b=*/false);

    // Extract ego coords -> voxel indices; trunc-toward-zero matches astype(int32)
    const int srcBase = lane & 16;
#pragma unroll
    for (int r = 0; r < 8; ++r) {
        float gx = acc[r];                               // N=0 at lanes 0 / 16
        float gy = __shfl(acc[r], srcBase + 1, 32);      // N=1
        float gz = __shfl(acc[r], srcBase + 2, 32);      // N=2
        if (m16 == 0) {
            int m  = r + (hi ? 8 : 0);
            int ix = (int)((gx + 51.2f) / 0.8f);
            int iy = (int)((gy + 51.2f) / 0.8f);
            int iz = (int)((gz + 10.0f) / 20.0f);
            bool valid = (ix >= 0) && (ix < NXV) && (iy >= 0) && (iy < NYV) && (iz == 0);
            voxIdx[pointBase + m] = valid ? (ix * NYV + iy) : -1;
        }
    }
}

// ---------------------------------------------------------------------------
// Fused softmax + run-compressed depth-weighted scatter.
// One 256-thread block per pixel (n,h,w). Consecutive depth bins that hit the
// same voxel are merged in LDS before issuing atomics (the dominant cost).
// ---------------------------------------------------------------------------
__global__ void scatter_kernel(const float* __restrict__ img_feat,
                               const float* __restrict__ depth_digit,
                               const int* __restrict__ voxIdx,
                               float* __restrict__ out) {
    __shared__ float probs[DBINS];
    __shared__ float feats[CCH];
    __shared__ int   vox[DBINS];
    __shared__ float red[128];
    __shared__ int   cVox[DBINS];
    __shared__ float cW[DBINS];
    __shared__ int   nRuns;

    int pix = blockIdx.x;
    int w = pix % WFD;  int t = pix / WFD;
    int h = t % HFD;    int n = t / HFD;
    int tid = threadIdx.x;
    const int hw = h * WFD + w;

    // ---- softmax over the 112 depth logits ----
    float logit = -3.0e38f;
    if (tid < DBINS) logit = depth_digit[(n * DBINS + tid) * HW + hw];
    if (tid < 128) red[tid] = (tid < DBINS) ? logit : -3.0e38f;
    __syncthreads();
    for (int s = 64; s > 0; s >>= 1) {
        if (tid < s) red[tid] = fmaxf(red[tid], red[tid + s]);
        __syncthreads();
    }
    float mx = red[0];
    __syncthreads();
    float e = 0.0f;
    if (tid < DBINS) e = expf(logit - mx);
    if (tid < 128) red[tid] = (tid < DBINS) ? e : 0.0f;
    __syncthreads();
    for (int s = 64; s > 0; s >>= 1) {
        if (tid < s) red[tid] += red[tid + s];
        __syncthreads();
    }
    float inv = 1.0f / red[0];

    // ---- stage per-pixel data in LDS ----
    if (tid == 0) nRuns = 0;
    if (tid < DBINS) {
        probs[tid] = e * inv;
        vox[tid]   = voxIdx[(n * DBINS + tid) * HW + hw];
    }
    if (tid < CCH) feats[tid] = img_feat[(n * CCH + tid) * HW + hw];
    __syncthreads();

    // ---- run-length merge consecutive depth bins with equal voxel index ----
    if (tid < DBINS) {
        int v = vox[tid];
        if (v >= 0 && (tid == 0 || vox[tid - 1] != v)) {
            float s = 0.0f;
            int dd = tid;
            while (dd < DBINS && vox[dd] == v) { s += probs[dd]; ++dd; }
            int slot = atomicAdd(&nRuns, 1);
            cVox[slot] = v;
            cW[slot]   = s;
        }
    }
    __syncthreads();

    // ---- scatter: nRuns * 80 atomics instead of 112 * 80 ----
    int nr = nRuns;
    for (int i = tid; i < nr * CCH; i += blockDim.x) {
        int rI = i / CCH;
        int c  = i - rI * CCH;
        atomicAdd(out + c * (NXV * NYV) + cVox[rI], cW[rI] * feats[c]);
    }
}

__global__ void zero_kernel(float4* __restrict__ out, int nvec) {
    int i = blockIdx.x * blockDim.x + threadIdx.x;
    if (i < nvec) out[i] = make_float4(0.0f, 0.0f, 0.0f, 0.0f);
}

// ---------------------------------------------------------------------------
// Inputs (setup_inputs order): 0:x 1:rots 2:trans 3:intrins 4:post_rots
//                              5:post_trans 6:img_feat 7:depth_digit
// ---------------------------------------------------------------------------
extern "C" void kernel_launch(void* const* d_in, const int* in_sizes, int n_in,
                              void* d_out, int out_size, void* d_ws, size_t ws_size,
                              hipStream_t stream) {
    const float* rots        = (const float*)d_in[1];
    const float* trans       = (const float*)d_in[2];
    const float* intrins     = (const float*)d_in[3];
    const float* post_rots   = (const float*)d_in[4];
    const float* post_trans  = (const float*)d_in[5];
    const float* img_feat    = (const float*)d_in[6];
    const float* depth_digit = (const float*)d_in[7];
    float* out = (float*)d_out;

    float* mats = (float*)d_ws;                      // 6*24 floats
    int* voxIdx = (int*)((char*)d_ws + 1024);        // 473088 ints (~1.9 MB)

    int nvec = out_size / 4;                         // out_size = 1310720, /4 exact
    zero_kernel<<<(nvec + 255) / 256, 256, 0, stream>>>((float4*)out, nvec);

    prep_kernel<<<1, 32, 0, stream>>>(rots, trans, intrins, post_rots, post_trans, mats);

    const int waves  = NPTS / 16;   // 29568 waves, 16 points each
    const int blocks = waves / 8;   // 256-thread blocks = 8 waves
    geom_kernel<<<blocks, 256, 0, stream>>>(mats, voxIdx);

    scatter_kernel<<<NCAM * HW, 256, 0, stream>>>(img_feat, depth_digit, voxIdx, out);
}